// GraphConvLayer_37941741093504
// MI455X (gfx1250) — compile-verified
//
#include <hip/hip_runtime.h>

typedef __attribute__((ext_vector_type(2))) float v2f;
typedef __attribute__((ext_vector_type(8))) float v8f;

#define FEAT 64

// ---------------- Stage 1: degree accumulation ----------------
__global__ void gc_degrees(const int* __restrict__ src, const int* __restrict__ dst,
                           float* __restrict__ deg_out, float* __restrict__ deg_in, int E) {
    int i = blockIdx.x * blockDim.x + threadIdx.x;
    if (i < E) {
        atomicAdd(&deg_out[src[i]], 1.0f);
        atomicAdd(&deg_in[dst[i]], 1.0f);
    }
}

// ---------------- Stage 2: norms (in place) ----------------
__global__ void gc_norms(float* __restrict__ ns, float* __restrict__ nd, int n) {
    int i = blockIdx.x * blockDim.x + threadIdx.x;
    if (i < n) {
        ns[i] = rsqrtf(fmaxf(ns[i], 1.0f));
        nd[i] = rsqrtf(fmaxf(nd[i], 1.0f));
    }
}

// ---------------- Stage 3: edge scatter (SpMM) ----------------
// One thread per (edge, feature-pair): 32 threads cover one edge's 64 features.
__global__ void gc_scatter(const float* __restrict__ x, const int* __restrict__ src,
                           const int* __restrict__ dst, const float* __restrict__ nsrc,
                           float* __restrict__ agg, int E) {
    int gid = blockIdx.x * blockDim.x + threadIdx.x;   // E*32 = 51.2M < 2^31
    if (gid >= E * 32) return;
    int e  = gid >> 5;
    int d2 = (gid & 31) * 2;
    int s = src[e];
    int t = dst[e];
    float ns = nsrc[s];
    v2f v = *(const v2f*)(x + (long long)s * FEAT + d2);   // 8B-aligned (d2 even)
    float* ap = agg + (long long)t * FEAT + d2;
    atomicAdd(ap,     v.x * ns);
    atomicAdd(ap + 1, v.y * ns);
}

// ---------------- Stage 4: WMMA GEMM: out = (agg*nd) @ W + b ----------------
// 256 threads = 8 waves; each wave owns 16 node-rows; W+bias staged in LDS.
__global__ __launch_bounds__(256) void gc_gemm(const float* __restrict__ agg,
                                               const float* __restrict__ ndst,
                                               const float* __restrict__ W,
                                               const float* __restrict__ bias,
                                               float* __restrict__ out, int n) {
    __shared__ float Wlds[FEAT * FEAT];
    __shared__ float blds[FEAT];
    for (int i = threadIdx.x; i < FEAT * FEAT; i += 256) Wlds[i] = W[i];
    if (threadIdx.x < FEAT) blds[threadIdx.x] = bias[threadIdx.x];
    __syncthreads();

    int wave = threadIdx.x >> 5;
    int lane = threadIdx.x & 31;
    int rowBase = (blockIdx.x * 8 + wave) * 16;
    if (rowBase >= n) return;                 // wave-uniform: EXEC all-1s inside

    int m     = lane & 15;                    // row within tile (A), col within tile (B/D)
    int khalf = (lane >> 4) * 2;              // lanes 0-15 -> K={0,1}; 16-31 -> K={2,3}
    int row   = rowBase + m;
    float nd  = ndst[row];

    // Preload all 16 A fragments for this wave's 16x64 tile, fusing norm_dst.
    v2f A[16];
    const float* hrow = agg + (long long)row * FEAT;
#pragma unroll
    for (int ks = 0; ks < 16; ++ks) {
        v2f t = *(const v2f*)(hrow + ks * 4 + khalf);
        A[ks].x = t.x * nd;
        A[ks].y = t.y * nd;
    }

    int mhalf = (lane >> 4) * 8;              // D-layout: lanes 16-31 hold rows M=8..15
#pragma unroll
    for (int nt = 0; nt < 4; ++nt) {
        int ncol = nt * 16 + m;
        v8f acc = {};
#pragma unroll
        for (int ks = 0; ks < 16; ++ks) {
            v2f B;
            B.x = Wlds[(ks * 4 + khalf) * FEAT + ncol];
            B.y = Wlds[(ks * 4 + khalf + 1) * FEAT + ncol];
            acc = __builtin_amdgcn_wmma_f32_16x16x4_f32(
                /*neg_a=*/false, A[ks], /*neg_b=*/false, B,
                /*c_mod=*/(short)0, acc, /*reuse_a=*/false, /*reuse_b=*/false);
        }
        float bv = blds[ncol];
#pragma unroll
        for (int r = 0; r < 8; ++r) {
            out[(long long)(rowBase + mhalf + r) * FEAT + ncol] = acc[r] + bv;
        }
    }
}

// ---------------- launcher ----------------
extern "C" void kernel_launch(void* const* d_in, const int* in_sizes, int n_in,
                              void* d_out, int out_size, void* d_ws, size_t ws_size,
                              hipStream_t stream) {
    const float* x    = (const float*)d_in[0];
    const int*   src  = (const int*)d_in[1];
    const int*   dst  = (const int*)d_in[2];
    const float* W    = (const float*)d_in[3];
    const float* bias = (const float*)d_in[4];
    float*       out  = (float*)d_out;

    const int n = in_sizes[0] / FEAT;   // 100000
    const int E = in_sizes[1];          // 1600000

    // workspace layout: norm_src[n] | norm_dst[n] | agg[n*64]
    float* nsrc = (float*)d_ws;
    float* ndst = nsrc + n;
    float* agg  = ndst + n;
    size_t used_bytes = (size_t)(2 + FEAT) * n * sizeof(float);

    hipMemsetAsync(d_ws, 0, used_bytes, stream);

    const int B = 256;
    gc_degrees<<<(E + B - 1) / B, B, 0, stream>>>(src, dst, nsrc, ndst, E);
    gc_norms<<<(n + B - 1) / B, B, 0, stream>>>(nsrc, ndst, n);

    int scatter_threads = E * 32;
    gc_scatter<<<(scatter_threads + B - 1) / B, B, 0, stream>>>(x, src, dst, nsrc, agg, E);

    int tiles = (n + 15) / 16;                  // 6250
    int gemm_blocks = (tiles + 7) / 8;          // 8 waves (16 rows each) per block
    gc_gemm<<<gemm_blocks, B, 0, stream>>>(agg, ndst, W, bias, out, n);
}